// TimeWarp_39221641347373
// MI455X (gfx1250) — compile-verified
//
#include <hip/hip_runtime.h>
#include <hip/hip_bf16.h>

typedef __attribute__((ext_vector_type(2))) float v2f;
typedef __attribute__((ext_vector_type(8))) float v8f;

#define B_   32
#define T_   4096
#define C_   128
#define K_   6
#define NSEG 5
#define H_   0.2f
#define JPAD 12   // 6 y-coeffs + 4 interior z-coeffs + 2 zero pad -> 3 WMMA k=4 steps

// Spline basis weight Phi[t, j] for coefficient vector [y0..y5, z1..z4, 0, 0].
// Branch-free (selects) so EXEC stays all-1s around the WMMA.
__device__ __forceinline__ float phi_basis(float tt, int j) {
    int i = (int)(tt * (float)NSEG);
    i = i > NSEG - 1 ? NSEG - 1 : i;
    float a = (float)(i + 1) * H_ - tt;   // ws[i+1] - t
    float b = tt - (float)i * H_;         // t - ws[i]
    const float invh = 1.0f / H_;
    const float c3   = 1.0f / (6.0f * H_);
    const float cl   = H_ / 6.0f;
    float v = 0.0f;
    v += (j == i)     ? a * invh : 0.0f;                                 // y_i weight
    v += (j == i + 1) ? b * invh : 0.0f;                                 // y_{i+1} weight
    v += (j == i + 5 && i >= 1) ? (a * a * a * c3 - a * cl) : 0.0f;      // z_i   (z0 dropped)
    v += (j == i + 6 && i <= 3) ? (b * b * b * c3 - b * cl) : 0.0f;      // z_{i+1} (z5 dropped)
    return v;
}

// ---- Setup: per (b,c) solve the 4x4 tridiagonal system, store coef + scale ----
__global__ void __launch_bounds__(256)
warp_setup(const float* __restrict__ warps, float* __restrict__ wsbuf) {
    int id = blockIdx.x * blockDim.x + threadIdx.x;   // id = b*C + c
    if (id >= B_ * C_) return;
    int b = id / C_, c = id % C_;

    float y[K_];
#pragma unroll
    for (int k = 0; k < K_; ++k)
        y[k] = ((float)k * H_) * warps[(b * K_ + k) * C_ + c];

    float rhs[4];
#pragma unroll
    for (int j = 0; j < 4; ++j)   // 6/h^2 = 150
        rhs[j] = 150.0f * (y[j + 2] - 2.0f * y[j + 1] + y[j]);

    const float s = 1.0f / 209.0f;  // 209*Minv hard-coded (tridiag [1,4,1] inverse)
    float z1 = ( 56.f*rhs[0] - 15.f*rhs[1] +  4.f*rhs[2] -  1.f*rhs[3]) * s;
    float z2 = (-15.f*rhs[0] + 60.f*rhs[1] - 16.f*rhs[2] +  4.f*rhs[3]) * s;
    float z3 = (  4.f*rhs[0] - 16.f*rhs[1] + 60.f*rhs[2] - 15.f*rhs[3]) * s;
    float z4 = ( -1.f*rhs[0] +  4.f*rhs[1] - 15.f*rhs[2] + 56.f*rhs[3]) * s;

    float* coef = wsbuf;                                    // [B][JPAD][C]
#pragma unroll
    for (int k = 0; k < K_; ++k) coef[(b * JPAD + k) * C_ + c] = y[k];
    coef[(b * JPAD +  6) * C_ + c] = z1;
    coef[(b * JPAD +  7) * C_ + c] = z2;
    coef[(b * JPAD +  8) * C_ + c] = z3;
    coef[(b * JPAD +  9) * C_ + c] = z4;
    coef[(b * JPAD + 10) * C_ + c] = 0.0f;
    coef[(b * JPAD + 11) * C_ + c] = 0.0f;

    float* scale = wsbuf + B_ * JPAD * C_;                  // [B][C]
    scale[b * C_ + c] = (float)(T_ - 1) / y[K_ - 1];        // tw at t=1 == y5 exactly
}

// ---- Main: one wave per 16x16 (t,c) tile; tw via 3x V_WMMA_F32_16X16X4_F32,
// then clip/searchsorted/gather/lerp straight out of the accumulator. ----
__global__ void __launch_bounds__(256)
warp_apply(const float* __restrict__ x, const float* __restrict__ wsbuf,
           float* __restrict__ out) {
    const int lane = threadIdx.x & 31;
    const int wave = threadIdx.x >> 5;
    int tile = blockIdx.x * 8 + wave;          // B*(T/16)*(C/16) = 65536 tiles
    int ct   = tile & 7;                       // C tile
    int tt   = (tile >> 3) & 255;              // T tile
    int b    = tile >> 11;                     // batch

    const float* coef     = wsbuf;
    const float* scaleArr = wsbuf + B_ * JPAD * C_;

    const int m     = lane & 15;               // A row (= output N col)
    const int khalf = (lane >> 4) * 2;         // K split across lane halves
    const float trow = (float)(tt * 16 + m) * (1.0f / (float)(T_ - 1));
    const int c = ct * 16 + m;

    v8f acc = {};
#pragma unroll
    for (int kc = 0; kc < 3; ++kc) {
        int j0 = 4 * kc + khalf;
        v2f afrag, bfrag;
        afrag.x = phi_basis(trow, j0);
        afrag.y = phi_basis(trow, j0 + 1);
        bfrag.x = coef[(b * JPAD + j0    ) * C_ + c];
        bfrag.y = coef[(b * JPAD + j0 + 1) * C_ + c];
        acc = __builtin_amdgcn_wmma_f32_16x16x4_f32(
            false, afrag, false, bfrag, (short)0, acc, false, false);
    }

    const float scl = scaleArr[b * C_ + c];
    const float* xb = x   + (size_t)b * T_ * C_ + c;
    float*       ob = out + (size_t)b * T_ * C_ + c;
    const int tbase = tt * 16 + ((lane >> 4) << 3);   // C/D: lanes 16-31 hold M=r+8

#pragma unroll
    for (int r = 0; r < 8; ++r) {
        float tw  = acc[r];
        float stw = fminf(fmaxf(scl * tw, 0.0f), (float)(T_ - 1));
        float q   = fminf(stw, 1.0f);                 // reference clips to orig[-1]=1.0
        float pos = q * (float)(T_ - 1);
        int   idx = (int)ceilf(pos);                  // searchsorted-left on uniform grid
        idx = idx < 1 ? 1 : idx;                      // clip low; q<=1 -> idx<=T-1
        float y0 = xb[(size_t)(idx - 1) * C_];
        float y1 = xb[(size_t)idx * C_];
        float f  = pos - (float)(idx - 1);
        ob[(size_t)(tbase + r) * C_] = fmaf(y1 - y0, f, y0);
    }
}

extern "C" void kernel_launch(void* const* d_in, const int* in_sizes, int n_in,
                              void* d_out, int out_size, void* d_ws, size_t ws_size,
                              hipStream_t stream) {
    const float* x     = (const float*)d_in[0];   // (B,T,C) f32
    const float* warps = (const float*)d_in[1];   // (B,K,C) f32
    float*       out   = (float*)d_out;           // (B,T,C) f32
    float*       wsbuf = (float*)d_ws;            // coef (B*12*C) + scale (B*C)

    warp_setup<<<(B_ * C_ + 255) / 256, 256, 0, stream>>>(warps, wsbuf);

    int tiles = B_ * (T_ / 16) * (C_ / 16);       // 65536, 8 waves/block
    warp_apply<<<tiles / 8, 256, 0, stream>>>(x, wsbuf, out);
}